// MCANet_6193342841275
// MI455X (gfx1250) — compile-verified
//
#include <hip/hip_runtime.h>
#include <hip/hip_bf16.h>

// MCANet on MI455X (gfx1250, wave32).
// Output = x * sigmoid(w1d[:,center] . zpad)  with z = mean_{h,w}(x).
// The conv1x1+softmax in the reference is dead code and is eliminated.
// Traffic-optimal: 1 HBM read of x (fills 192MB L2; x=128MB), tiny WMMA gate
// kernel, then L2-resident re-read + NT stores for the output.
// Roofline: 256MB min HBM traffic at 23.3 TB/s ~= 11 us.

typedef __attribute__((ext_vector_type(2))) float v2f;
typedef __attribute__((ext_vector_type(4))) float v4f;
typedef __attribute__((ext_vector_type(8))) float v8f;

#define BATCH  16
#define CCH    512
#define HWSZ   4096   // 64*64
#define KTAPS  5
#define CENTER 2      // (5-1)/2

// ---------------------------------------------------------------------------
// Pass 1: z[b,c] = mean over H*W of x[b,c,:,:]. One block per (b,c).
// Coalesced 128-bit loads (global_load_b128), wave32 shuffle reduce, LDS tree.
// ---------------------------------------------------------------------------
__global__ __launch_bounds__(256) void mca_mean_kernel(const float* __restrict__ x,
                                                       float* __restrict__ z) {
    const int bc = blockIdx.x;
    const v4f* p = (const v4f*)(x + (size_t)bc * HWSZ);
    const int t = threadIdx.x;

    float s = 0.0f;
#pragma unroll
    for (int i = 0; i < 4; ++i) {
        v4f v = p[t + i * 256];
        s += (v.x + v.y) + (v.z + v.w);
    }
    // wave32 butterfly reduce
#pragma unroll
    for (int off = 16; off >= 1; off >>= 1)
        s += __shfl_xor(s, off, 32);

    __shared__ float wsum[8];
    if ((t & 31) == 0) wsum[t >> 5] = s;
    __syncthreads();
    if (t == 0) {
        float tot = 0.0f;
#pragma unroll
        for (int i = 0; i < 8; ++i) tot += wsum[i];
        z[bc] = tot * (1.0f / (float)HWSZ);
    }
}

// ---------------------------------------------------------------------------
// Gate kernel: per 16-channel tile, D = A x B via V_WMMA_F32_16X16X4_F32 where
//   A[m][k] = zpad[c0+m][k]   (nonzero only at k==CENTER)
//   B[k][n] = w1d[c0+n][k]
// diag(D)[m] = sum_k zpad[c0+m][k]*w1d[c0+m][k]  == the reference's full dot
// against the padded window. Two K=4 WMMAs cover taps 0..7 (5..7 zero-pad).
// One wave (32 lanes, EXEC all-ones) per tile; 16*32 = 512 tiles.
//
// A 16x4 f32 layout (ISA 7.12.2): lanes 0-15 hold M=lane, vgpr0=K0, vgpr1=K1;
// lanes 16-31 hold M=lane-16, vgpr0=K2, vgpr1=K3. B 4x16 mirrors it.
// C/D 16x16 f32: vgpr v = row M=v (lanes 0-15, N=lane) / M=v+8 (lanes 16-31).
// ---------------------------------------------------------------------------
__global__ __launch_bounds__(32) void mca_gate_wmma_kernel(const float* __restrict__ z,
                                                           const float* __restrict__ w1d,
                                                           float* __restrict__ gate) {
    const int b    = blockIdx.x >> 5;          // 16 batches
    const int c0   = (blockIdx.x & 31) << 4;   // 32 tiles of 16 channels
    const int lane = threadIdx.x;
    const int row  = lane & 15;
    const int hi   = lane >> 4;                // 0: K={0,1}, 1: K={2,3}
    const int c    = c0 + row;

    // A fragments (zpad): only tap CENTER==2 is nonzero -> vgpr0 of hi half.
    v2f a1, a2;
    a1.x = hi ? z[b * CCH + c] : 0.0f;   // K = hi?2:0
    a1.y = 0.0f;                         // K = hi?3:1
    a2.x = 0.0f;                         // K = hi?6:4  (zpad[,4..7] == 0)
    a2.y = 0.0f;                         // K = hi?7:5

    // B fragments: B[k][n] = w1d[c0+n][k]
    const float* wr = w1d + (size_t)c * KTAPS;
    v2f bf1, bf2;
    bf1.x = wr[hi ? 2 : 0];
    bf1.y = wr[hi ? 3 : 1];
    bf2.x = hi ? 0.0f : wr[4];           // only tap 4 exists beyond K=4
    bf2.y = 0.0f;

    v8f acc = {};
    acc = __builtin_amdgcn_wmma_f32_16x16x4_f32(false, a1, false, bf1,
                                                (short)0, acc, false, false);
    acc = __builtin_amdgcn_wmma_f32_16x16x4_f32(false, a2, false, bf2,
                                                (short)0, acc, false, false);

    // Diagonal extraction: M=0..7 -> lane m, vgpr m ; M=8..15 -> lane m+16, vgpr m-8.
    const bool writer = (lane < 8) || (lane >= 24);
    const int  m      = (lane < 8) ? lane : (lane - 16);
    const int  idx    = (lane < 8) ? lane : (lane - 24);
    float zc = 0.0f;
#pragma unroll
    for (int i = 0; i < 8; ++i) zc = (idx == i) ? acc[i] : zc;

    if (writer) {
        float g = 1.0f / (1.0f + __expf(-zc));
        gate[b * CCH + c0 + m] = g;
    }
}

// ---------------------------------------------------------------------------
// Pass 2: out = x * gate[b,c]. x re-read should hit L2 (128MB resident in
// 192MB L2); output stored non-temporally (TH=NT) so it doesn't evict x.
// ---------------------------------------------------------------------------
__global__ __launch_bounds__(256) void mca_scale_kernel(const float* __restrict__ x,
                                                        const float* __restrict__ gate,
                                                        float* __restrict__ out) {
    const int bc = blockIdx.x;
    const float g = gate[bc];    // uniform per block -> scalar load
    const v4f* px = (const v4f*)(x + (size_t)bc * HWSZ);
    v4f* po = (v4f*)(out + (size_t)bc * HWSZ);
    const int t = threadIdx.x;
#pragma unroll
    for (int i = 0; i < 4; ++i) {
        v4f v = px[t + i * 256];
        v *= g;
        __builtin_nontemporal_store(v, po + t + i * 256);
    }
}

// ---------------------------------------------------------------------------
extern "C" void kernel_launch(void* const* d_in, const int* in_sizes, int n_in,
                              void* d_out, int out_size, void* d_ws, size_t ws_size,
                              hipStream_t stream) {
    const float* x   = (const float*)d_in[0];  // (16,512,64,64)
    // d_in[1] = w1x1, d_in[2] = b1x1 : dead code in reference, never read.
    const float* w1d = (const float*)d_in[3];  // (512,5)
    float* out = (float*)d_out;

    float* z    = (float*)d_ws;          // 16*512 floats
    float* gate = z + BATCH * CCH;       // 16*512 floats

    const int nbc = BATCH * CCH;         // 8192

    mca_mean_kernel<<<nbc, 256, 0, stream>>>(x, z);
    mca_gate_wmma_kernel<<<BATCH * 32, 32, 0, stream>>>(z, w1d, gate);
    mca_scale_kernel<<<nbc, 256, 0, stream>>>(x, gate, out);
}